// GAT_50276887167406
// MI455X (gfx1250) — compile-verified
//
#include <hip/hip_runtime.h>
#include <math.h>

// ---------------- problem constants (match reference) ----------------
#define N_NODES 20000
#define N_EDGES 320000
#define ETOT    (N_EDGES + N_NODES)   // self-loops appended
#define F_IN    128
#define H1      8
#define C1      64
#define D1      512                    // H1*C1
#define NC      10
#define NGRAPH  64
#define NEG_SLOPE 0.2f

typedef __attribute__((ext_vector_type(2))) float v2f;
typedef __attribute__((ext_vector_type(8))) float v8f;

// ---------------- helpers ----------------
__device__ __forceinline__ float atomicMaxFloat(float* addr, float v) {
  // standard two-sided trick: works with -inf initialization
  if (v >= 0.0f)
    return __int_as_float(atomicMax((int*)addr, __float_as_int(v)));
  else
    return __uint_as_float(atomicMin((unsigned int*)addr, __float_as_uint(v)));
}

__device__ __forceinline__ void edge_ends(const int* __restrict__ ei, int e, int& s, int& d) {
  if (e < N_EDGES) { s = ei[e]; d = ei[N_EDGES + e]; }
  else             { s = e - N_EDGES; d = s; }        // self loop
}

// ---------------- init ----------------
__global__ void fill_kernel(float* __restrict__ p, int n, float v) {
  for (int i = blockIdx.x * blockDim.x + threadIdx.x; i < n; i += gridDim.x * blockDim.x)
    p[i] = v;
}

// ---------------- GEMM1: h1 = x @ W1  via V_WMMA_F32_16X16X4_F32 ----------------
// grid = 1250 blocks (one 16-row stripe each), 256 threads = 8 waves, wave covers 64 cols
__global__ __launch_bounds__(256) void gemm1_wmma(const float* __restrict__ X,
                                                  const float* __restrict__ W,
                                                  float* __restrict__ Hout) {
  const int wave   = threadIdx.x >> 5;
  const int lane   = threadIdx.x & 31;
  const int laneM  = lane & 15;
  const int half   = lane >> 4;          // 0 -> K pair {0,1}, 1 -> K pair {2,3}
  const int mBase  = blockIdx.x * 16;
  const int colBase = wave * 64;

  v8f acc0 = {}, acc1 = {}, acc2 = {}, acc3 = {};
  const float* xrow = X + (size_t)(mBase + laneM) * F_IN;

#pragma unroll 4
  for (int k = 0; k < F_IN; k += 4) {
    const int ka = k + 2 * half;
    v2f a; a.x = xrow[ka]; a.y = xrow[ka + 1];

    const float* w0 = W + (size_t)ka * D1 + colBase + laneM;
    const float* w1 = w0 + D1;
    __builtin_prefetch(w0 + 4 * D1, 0, 0);    // global_prefetch_b8
    v2f b0 = { w0[0],  w1[0]  };
    v2f b1 = { w0[16], w1[16] };
    v2f b2 = { w0[32], w1[32] };
    v2f b3 = { w0[48], w1[48] };

    acc0 = __builtin_amdgcn_wmma_f32_16x16x4_f32(false, a, false, b0, (short)0, acc0, false, false);
    acc1 = __builtin_amdgcn_wmma_f32_16x16x4_f32(false, a, false, b1, (short)0, acc1, false, false);
    acc2 = __builtin_amdgcn_wmma_f32_16x16x4_f32(false, a, false, b2, (short)0, acc2, false, false);
    acc3 = __builtin_amdgcn_wmma_f32_16x16x4_f32(false, a, false, b3, (short)0, acc3, false, false);
  }

#pragma unroll
  for (int r = 0; r < 8; ++r) {
    const int row = mBase + r + 8 * half;           // C/D layout: VGPR r -> M=r (lanes 0-15), M=r+8 (lanes 16-31)
    float* o = Hout + (size_t)row * D1 + colBase + laneM;
    o[0]  = acc0[r];
    o[16] = acc1[r];
    o[32] = acc2[r];
    o[48] = acc3[r];
  }
}

// ---------------- GEMM2: g2 = helu @ W2 (N padded 10 -> 16) ----------------
__global__ __launch_bounds__(256) void gemm2_wmma(const float* __restrict__ Hin,
                                                  const float* __restrict__ W2,
                                                  float* __restrict__ Gout) {
  const int wave  = threadIdx.x >> 5;
  const int lane  = threadIdx.x & 31;
  const int laneM = lane & 15;
  const int half  = lane >> 4;
  const int tile  = blockIdx.x * 8 + wave;
  if (tile >= N_NODES / 16) return;                 // whole-wave uniform exit (EXEC stays full)
  const int mBase = tile * 16;

  v8f acc = {};
  const float* hrow = Hin + (size_t)(mBase + laneM) * D1;
  const bool colOK = (laneM < NC);

#pragma unroll 8
  for (int k = 0; k < D1; k += 4) {
    const int ka = k + 2 * half;
    v2f a; a.x = hrow[ka]; a.y = hrow[ka + 1];
    __builtin_prefetch(hrow + ka + 64, 0, 0);
    v2f b;
    b.x = colOK ? W2[(size_t)ka * NC + laneM] : 0.0f;
    b.y = colOK ? W2[(size_t)(ka + 1) * NC + laneM] : 0.0f;
    acc = __builtin_amdgcn_wmma_f32_16x16x4_f32(false, a, false, b, (short)0, acc, false, false);
  }

  if (colOK) {
#pragma unroll
    for (int r = 0; r < 8; ++r) {
      const int row = mBase + r + 8 * half;
      Gout[(size_t)row * NC + laneM] = acc[r];
    }
  }
}

// ---------------- attention logits per node ----------------
__global__ void att1_kernel(const float* __restrict__ Hm, const float* __restrict__ a_src,
                            const float* __restrict__ a_dst, float* __restrict__ As,
                            float* __restrict__ Ad) {
  const int idx = blockIdx.x * blockDim.x + threadIdx.x;     // n*8 + head
  if (idx >= N_NODES * H1) return;
  const int n = idx >> 3, hd = idx & 7;
  const float* h  = Hm + (size_t)n * D1 + hd * C1;
  const float* as = a_src + hd * C1;
  const float* ad = a_dst + hd * C1;
  float s = 0.f, d = 0.f;
#pragma unroll 8
  for (int c = 0; c < C1; ++c) { const float v = h[c]; s = fmaf(v, as[c], s); d = fmaf(v, ad[c], d); }
  As[idx] = s; Ad[idx] = d;
}

__global__ void att2_kernel(const float* __restrict__ Gm, const float* __restrict__ a_src,
                            const float* __restrict__ a_dst, float* __restrict__ As,
                            float* __restrict__ Ad) {
  const int n = blockIdx.x * blockDim.x + threadIdx.x;
  if (n >= N_NODES) return;
  const float* g = Gm + (size_t)n * NC;
  float s = 0.f, d = 0.f;
#pragma unroll
  for (int c = 0; c < NC; ++c) { const float v = g[c]; s = fmaf(v, a_src[c], s); d = fmaf(v, a_dst[c], d); }
  As[n] = s; Ad[n] = d;
}

// ---------------- edge passes (segment max, then exp + denom) ----------------
template <int H>
__global__ void edge_max_kernel(const int* __restrict__ ei, const float* __restrict__ As,
                                const float* __restrict__ Ad, float* __restrict__ Emax) {
  const int idx = blockIdx.x * blockDim.x + threadIdx.x;
  if (idx >= ETOT * H) return;
  const int e = idx / H, hd = idx % H;
  int s, d; edge_ends(ei, e, s, d);
  float v = As[s * H + hd] + Ad[d * H + hd];
  v = (v > 0.f) ? v : NEG_SLOPE * v;                 // leaky relu
  atomicMaxFloat(&Emax[d * H + hd], v);
}

template <int H>
__global__ void edge_exp_kernel(const int* __restrict__ ei, const float* __restrict__ As,
                                const float* __restrict__ Ad, const float* __restrict__ Emax,
                                float* __restrict__ EE, float* __restrict__ Den) {
  const int idx = blockIdx.x * blockDim.x + threadIdx.x;
  if (idx >= ETOT * H) return;
  const int e = idx / H, hd = idx % H;
  int s, d; edge_ends(ei, e, s, d);
  float v = As[s * H + hd] + Ad[d * H + hd];
  v = (v > 0.f) ? v : NEG_SLOPE * v;
  const float ee = __expf(v - Emax[d * H + hd]);
  EE[idx] = ee;                                      // idx == e*H + hd
  atomicAdd(&Den[d * H + hd], ee);
}

// ---------------- edge feature aggregation ----------------
// layer 1: one wave per edge, each lane handles 16 consecutive features (lane*16 stays in one head)
__global__ __launch_bounds__(256) void edge_agg1_kernel(const int* __restrict__ ei,
                                                        const float* __restrict__ Hm,
                                                        const float* __restrict__ EE,
                                                        float* __restrict__ Agg) {
  const int gid  = blockIdx.x * blockDim.x + threadIdx.x;
  const int e    = gid >> 5;
  const int lane = gid & 31;
  if (e >= ETOT) return;
  int s, d; edge_ends(ei, e, s, d);
  const int f0 = lane * 16;
  const float w = EE[e * H1 + (f0 >> 6)];
  const float4* hp = (const float4*)(Hm + (size_t)s * D1 + f0);
  float* ap = Agg + (size_t)d * D1 + f0;
#pragma unroll
  for (int j = 0; j < 4; ++j) {
    const float4 hv = hp[j];
    atomicAdd(&ap[j * 4 + 0], hv.x * w);
    atomicAdd(&ap[j * 4 + 1], hv.y * w);
    atomicAdd(&ap[j * 4 + 2], hv.z * w);
    atomicAdd(&ap[j * 4 + 3], hv.w * w);
  }
}

__global__ void edge_agg2_kernel(const int* __restrict__ ei, const float* __restrict__ Gm,
                                 const float* __restrict__ EE, float* __restrict__ Agg) {
  const int idx = blockIdx.x * blockDim.x + threadIdx.x;
  if (idx >= ETOT * NC) return;
  const int e = idx / NC, c = idx % NC;
  int s, d; edge_ends(ei, e, s, d);
  atomicAdd(&Agg[(size_t)d * NC + c], Gm[(size_t)s * NC + c] * EE[e]);
}

// ---------------- finalize layer 1: divide by denom, +bias, ELU (in place) ----------------
__global__ void fin1_kernel(const float* __restrict__ Den, const float* __restrict__ bias,
                            float* __restrict__ Agg) {
  const int idx = blockIdx.x * blockDim.x + threadIdx.x;
  if (idx >= N_NODES * D1) return;
  const int n = idx >> 9, f = idx & 511;
  const float v = Agg[idx] / (Den[n * H1 + (f >> 6)] + 1e-16f) + bias[f];
  Agg[idx] = (v > 0.f) ? v : (__expf(v) - 1.f);      // ELU(alpha=1)
}

// ---------------- finalize layer 2 + mean-pool accumulate ----------------
__global__ void fin2_kernel(const float* __restrict__ Den, const float* __restrict__ bias,
                            const int* __restrict__ batch, const float* __restrict__ Agg2,
                            float* __restrict__ pool, float* __restrict__ cnt) {
  const int idx = blockIdx.x * blockDim.x + threadIdx.x;
  if (idx >= N_NODES * NC) return;
  const int n = idx / NC, c = idx % NC;
  const float v = Agg2[idx] / (Den[n] + 1e-16f) + bias[c];
  const int g = batch[n];
  atomicAdd(&pool[g * NC + c], v);
  if (c == 0) atomicAdd(&cnt[g], 1.0f);
}

__global__ void pool_out_kernel(const float* __restrict__ pool, const float* __restrict__ cnt,
                                float* __restrict__ out) {
  const int i = blockIdx.x * blockDim.x + threadIdx.x;
  if (i >= NGRAPH * NC) return;
  out[i] = pool[i] / fmaxf(cnt[i / NC], 1.0f);
}

// ---------------- launch ----------------
static inline int cdiv(int a, int b) { return (a + b - 1) / b; }

extern "C" void kernel_launch(void* const* d_in, const int* in_sizes, int n_in,
                              void* d_out, int out_size, void* d_ws, size_t ws_size,
                              hipStream_t stream) {
  (void)in_sizes; (void)n_in; (void)out_size; (void)ws_size;

  const float* x      = (const float*)d_in[0];
  const int*   ei     = (const int*)  d_in[1];   // [2, E]
  const int*   batch  = (const int*)  d_in[2];
  const float* W1     = (const float*)d_in[3];
  const float* a_src1 = (const float*)d_in[4];
  const float* a_dst1 = (const float*)d_in[5];
  const float* b1     = (const float*)d_in[6];
  const float* W2     = (const float*)d_in[7];
  const float* a_src2 = (const float*)d_in[8];
  const float* a_dst2 = (const float*)d_in[9];
  const float* b2     = (const float*)d_in[10];
  float*       out    = (float*)d_out;

  // workspace layout (floats)
  float* p = (float*)d_ws;
  float* h1     = p; p += (size_t)N_NODES * D1;   // 10.24M
  float* agg1   = p; p += (size_t)N_NODES * D1;   // 10.24M  (becomes helu in place)
  float* asrc1  = p; p += N_NODES * H1;
  float* adst1  = p; p += N_NODES * H1;
  float* emax1  = p; p += N_NODES * H1;
  float* den1   = p; p += N_NODES * H1;
  float* ee1    = p; p += (size_t)ETOT * H1;      // 2.72M
  float* g2     = p; p += N_NODES * NC;
  float* agg2   = p; p += N_NODES * NC;
  float* asrc2  = p; p += N_NODES;
  float* adst2  = p; p += N_NODES;
  float* emax2  = p; p += N_NODES;
  float* den2   = p; p += N_NODES;
  float* ee2    = p; p += ETOT;
  float* pool   = p; p += NGRAPH * NC;
  float* cnt    = p; p += NGRAPH;

  const float NEGINF = -__builtin_huge_valf();
  // init accumulators (every call: harness does not re-clear between replays)
  fill_kernel<<<2048, 256, 0, stream>>>(agg1, N_NODES * D1, 0.0f);
  fill_kernel<<<64,   256, 0, stream>>>(den1, N_NODES * H1, 0.0f);
  fill_kernel<<<64,   256, 0, stream>>>(emax1, N_NODES * H1, NEGINF);
  fill_kernel<<<64,   256, 0, stream>>>(agg2, N_NODES * NC, 0.0f);
  fill_kernel<<<16,   256, 0, stream>>>(den2, N_NODES, 0.0f);
  fill_kernel<<<16,   256, 0, stream>>>(emax2, N_NODES, NEGINF);
  fill_kernel<<<4,    256, 0, stream>>>(pool, NGRAPH * NC + NGRAPH, 0.0f); // pool + cnt contiguous

  // ---- layer 1 ----
  gemm1_wmma<<<N_NODES / 16, 256, 0, stream>>>(x, W1, h1);
  att1_kernel<<<cdiv(N_NODES * H1, 256), 256, 0, stream>>>(h1, a_src1, a_dst1, asrc1, adst1);
  edge_max_kernel<H1><<<cdiv(ETOT * H1, 256), 256, 0, stream>>>(ei, asrc1, adst1, emax1);
  edge_exp_kernel<H1><<<cdiv(ETOT * H1, 256), 256, 0, stream>>>(ei, asrc1, adst1, emax1, ee1, den1);
  edge_agg1_kernel<<<cdiv(ETOT * 32, 256), 256, 0, stream>>>(ei, h1, ee1, agg1);
  fin1_kernel<<<cdiv(N_NODES * D1, 256), 256, 0, stream>>>(den1, b1, agg1);   // agg1 -> helu

  // ---- layer 2 ----
  gemm2_wmma<<<cdiv(N_NODES / 16, 8), 256, 0, stream>>>(agg1, W2, g2);
  att2_kernel<<<cdiv(N_NODES, 256), 256, 0, stream>>>(g2, a_src2, a_dst2, asrc2, adst2);
  edge_max_kernel<1><<<cdiv(ETOT, 256), 256, 0, stream>>>(ei, asrc2, adst2, emax2);
  edge_exp_kernel<1><<<cdiv(ETOT, 256), 256, 0, stream>>>(ei, asrc2, adst2, emax2, ee2, den2);
  edge_agg2_kernel<<<cdiv(ETOT * NC, 256), 256, 0, stream>>>(ei, g2, ee2, agg2);
  fin2_kernel<<<cdiv(N_NODES * NC, 256), 256, 0, stream>>>(den2, b2, batch, agg2, pool, cnt);

  // ---- mean pooling output ----
  pool_out_kernel<<<cdiv(NGRAPH * NC, 256), 256, 0, stream>>>(pool, cnt, out);
}